// LSTMEncoder_44916767981634
// MI455X (gfx1250) — compile-verified
//
#include <hip/hip_runtime.h>
#include <hip/hip_bf16.h>

// ---------------- problem constants (from reference) ----------------
#define Bc    32          // batch (the SCAN axis in the reference!)
#define Tc    2048        // rows: independent LSTM lanes
#define Dc    256         // input feature dim
#define Hc    256         // hidden dim per direction
#define Nc    8           // H / CS
#define CSc   32          // chunk size
#define Gc    1040        // 4*H + 2*N gates
#define GP    1152        // Gc padded to 72 N-tiles (8 waves x 9 tiles, no guards)
#define KTOT  512         // D + H (fused GEMM K)
#define MT    16          // rows per block = WMMA M tile
#define KC    16          // K-slice of W_cat staged in LDS per round
#define NTILES 65         // live tiles (Gc / 16)
#define WAVES 8
#define BLOCK 256
#define MAX_Q 9           // tiles per wave (padded, exact)

typedef __attribute__((ext_vector_type(2))) float v2f;
typedef __attribute__((ext_vector_type(4))) float v4f;
typedef __attribute__((ext_vector_type(8))) float v8f;

__device__ __forceinline__ float sigf(float v) {
    return 1.0f / (1.0f + __expf(-v));
}

// One block = 16 rows (one direction). Loops the 32 scan steps privately:
//   gates = [x[b,rows,:] | h] (16x512) x W_cat^T (512x1040) via v_wmma_f32_16x16x4_f32,
//   then ON-LSTM cell elementwise; h, c live in LDS across steps.
__global__ __launch_bounds__(BLOCK, 1)
void onlstm_fused_kernel(const float* __restrict__ x,
                         const long long* __restrict__ seq_lens,  // np.int64
                         const float* __restrict__ W_ih,
                         const float* __restrict__ b_ih,
                         const float* __restrict__ W_hh,
                         const float* __restrict__ b_hh,
                         float* __restrict__ out)
{
    extern __shared__ float lds[];
    float* sA = lds;                    // MT x KTOT      : A tile ([x | h] per row)
    float* sW = sA + MT * KTOT;         // (KC/2) x GP x 2: staged W slice, K-pair interleaved
    float* sG = sW + KC * GP;           // MT x Gc        : gates
    float* sC = sG + MT * Gc;           // MT x Hc        : cell state

    const int dir  = blockIdx.x & 1;            // 0 = fwd, 1 = feature-reversed
    const int t0   = (blockIdx.x >> 1) * MT;    // first row of this tile
    const int tid  = threadIdx.x;
    const int wave = tid >> 5;
    const int lane = tid & 31;
    const int ln   = lane & 15;                 // 0..15
    const int hi   = lane >> 4;                 // 0/1: K-pair select (WMMA f32 layout)

    // h0 = 0, c0 = 0; zero the whole sW once so the padded columns (g >= Gc)
    // stay 0 forever (staging only ever rewrites g < Gc).
    for (int i = tid; i < MT * Hc; i += BLOCK) {
        sA[(i >> 8) * KTOT + Hc + (i & 255)] = 0.0f;
        sC[i] = 0.0f;
    }
    for (int i = tid; i < KC * GP; i += BLOCK)
        sW[i] = 0.0f;

    // step-invariant bias per owned tile (padding tiles get 0 / never stored)
    float bias[MAX_Q];
#pragma unroll
    for (int q = 0; q < MAX_Q; ++q) {
        int nt = wave + q * WAVES;
        bias[q] = (nt < NTILES) ? (b_ih[nt * 16 + ln] + b_hh[nt * 16 + ln]) : 0.0f;
    }

    int outOff = 0;                             // prefix sum of seq_lens[0..b)
    for (int b = 0; b < Bc; ++b) {
        __syncthreads();                        // prev step's h/c/gates consumed

        // ---- load x[b, t0:t0+16, :] into sA[:, 0:256] as b128; for dir=1 the
        //      features are stored reversed (swizzled b128 store).
        for (int i = tid; i < MT * (Dc / 4); i += BLOCK) {   // 1024 items, exact
            int r  = i >> 6;                    // Dc/4 = 64 quads per row
            int d4 = (i & 63) << 2;
            v4f v = *(const v4f*)(x + (size_t)b * Tc * Dc +
                                      (size_t)(t0 + r) * Dc + d4);
            if (dir == 0) {
                *(v4f*)(sA + r * KTOT + d4) = v;
            } else {
                *(v4f*)(sA + r * KTOT + (Dc - 4 - d4)) = v.wzyx;
            }
        }
        // prefetch next step's x tile into L2 (global_prefetch_b8)
        if (b + 1 < Bc) {
            __builtin_prefetch(
                x + (size_t)(b + 1) * Tc * Dc + (size_t)(t0 + (tid >> 4)) * Dc +
                    ((tid & 15) << 4), 0, 0);
        }

        // ---- accumulators: wave w owns N-tiles w, w+8, ..., w+64 (exactly 9)
        v8f acc[MAX_Q];
#pragma unroll
        for (int q = 0; q < MAX_Q; ++q)
            acc[q] = v8f{0.f, 0.f, 0.f, 0.f, 0.f, 0.f, 0.f, 0.f};

        // ---- GEMM: K = 512 in KC slices staged through LDS
        for (int kc = 0; kc < KTOT; kc += KC) {
            __syncthreads();                    // sW free to overwrite
            // stage W_cat[:, kc:kc+KC): b128 global loads (4 consecutive k per
            // row g), two aligned b64 stores into the pair-interleaved layout.
            for (int i = tid; i < (KC / 4) * Gc; i += BLOCK) {   // 4160 items
                int k4i = i / Gc;               // 0..3
                int g   = i - k4i * Gc;
                int kl  = k4i << 2;             // 0,4,8,12
                int k   = kc + kl;              // slice never straddles Dc
                v4f w;
                if (k < Dc) {
                    if (dir == 0) {
                        w = *(const v4f*)(W_ih + g * Dc + k);
                    } else {
                        v4f t = *(const v4f*)(W_ih + g * Dc + (Dc - 4 - k));
                        w = t.wzyx;
                    }
                } else {
                    w = *(const v4f*)(W_hh + g * Dc + (k - Dc));
                }
                int kp = kl >> 1;               // even pair index
                *(v2f*)(sW + (kp    ) * (2 * GP) + 2 * g) = v2f{w.x, w.y};
                *(v2f*)(sW + (kp + 1) * (2 * GP) + 2 * g) = v2f{w.z, w.w};
            }
            __syncthreads();

#pragma unroll
            for (int k4 = 0; k4 < KC; k4 += 4) {
                // A frag (16x4 f32): lane<16 -> K=k4,k4+1 ; lane>=16 -> K=k4+2,k4+3
                const v2f a = *(const v2f*)(sA + ln * KTOT + kc + k4 + 2 * hi);
                const float* wb = sW + ((k4 >> 1) + hi) * (2 * GP) + 2 * ln;
                // load ALL 9 B fragments first (distinct regs -> DS loads stay
                // in flight, graded s_wait_dscnt before each wmma group)
                v2f bf[MAX_Q];
#pragma unroll
                for (int q = 0; q < MAX_Q; ++q)
                    bf[q] = *(const v2f*)(wb + 2 * ((wave + q * WAVES) * 16));
#pragma unroll
                for (int q = 0; q < MAX_Q; ++q)
                    acc[q] = __builtin_amdgcn_wmma_f32_16x16x4_f32(
                        false, a, false, bf[q], (short)0, acc[q], false, false);
            }
        }

        // ---- spill live gates (+bias) to LDS.  C/D layout: VGPR r = row r / r+8
#pragma unroll
        for (int q = 0; q < MAX_Q; ++q) {
            int nt = wave + q * WAVES;
            if (nt < NTILES) {                  // only padding tiles skipped
                int g = nt * 16 + ln;
#pragma unroll
                for (int r = 0; r < 8; ++r)
                    sG[(r + 8 * hi) * Gc + g] = acc[q][r] + bias[q];
            }
        }
        __syncthreads();

        // ---- ON-LSTM cell elementwise: 16 threads per row, 16 h-elements each
        const int row = tid >> 4;
        const int tr  = tid & 15;
        const long long sl = seq_lens[b];

        // cumsoftmax over gates[0:8] and gates[8:16] (redundant across the 16
        // threads of a row; only 8-wide, cheap VALU)
        float cin[Nc], cfg[Nc];
        {
            float e[Nc];
            float m = -3.0e38f;
#pragma unroll
            for (int n = 0; n < Nc; ++n) { e[n] = sG[row * Gc + n]; m = fmaxf(m, e[n]); }
            float s = 0.f;
#pragma unroll
            for (int n = 0; n < Nc; ++n) { e[n] = __expf(e[n] - m); s += e[n]; }
            float inv = 1.0f / s, c = 0.f;
#pragma unroll
            for (int n = 0; n < Nc; ++n) { c += e[n]; cin[n] = 1.0f - c * inv; }

            m = -3.0e38f;
#pragma unroll
            for (int n = 0; n < Nc; ++n) { e[n] = sG[row * Gc + Nc + n]; m = fmaxf(m, e[n]); }
            s = 0.f;
#pragma unroll
            for (int n = 0; n < Nc; ++n) { e[n] = __expf(e[n] - m); s += e[n]; }
            inv = 1.0f / s; c = 0.f;
#pragma unroll
            for (int n = 0; n < Nc; ++n) { c += e[n]; cfg[n] = c * inv; }
        }

        const int tglob = t0 + row;
        const bool wr = (long long)tglob < sl;
        const size_t obase = (size_t)(outOff + tglob) * (2 * Hc) + (size_t)dir * Hc;

#pragma unroll
        for (int j = 0; j < 16; ++j) {
            int sidx = tr * 16 + j;             // 0..255 (= n*CS + s)
            int n = sidx >> 5;
            int s = sidx & 31;
            float og = sG[row * Gc +  16 + n * CSc + s];
            float cg = sG[row * Gc + 272 + n * CSc + s];
            float ig = sG[row * Gc + 528 + n * CSc + s];
            float fg = sG[row * Gc + 784 + n * CSc + s];
            float ov  = cfg[n] * cin[n];
            float fgv = sigf(fg) * ov + (cfg[n] - ov);
            float igv = sigf(ig) * ov + (cin[n] - ov);
            float cx  = sC[row * Hc + sidx];
            float cy  = fgv * cx + igv * tanhf(cg);
            float hy  = sigf(og) * tanhf(cy);
            sC[row * Hc + sidx] = cy;
            sA[row * KTOT + Hc + sidx] = hy;    // next step's A operand (h part)
            if (wr) out[obase + sidx] = hy;
        }
        outOff += (int)sl;
    }
}

extern "C" void kernel_launch(void* const* d_in, const int* in_sizes, int n_in,
                              void* d_out, int out_size, void* d_ws, size_t ws_size,
                              hipStream_t stream) {
    const float*     x    = (const float*)d_in[0];
    const long long* sl   = (const long long*)d_in[1];   // seq_lens: np.int64
    const float*     W_ih = (const float*)d_in[2];
    const float*     b_ih = (const float*)d_in[3];
    const float*     W_hh = (const float*)d_in[4];
    const float*     b_hh = (const float*)d_in[5];
    float*           out  = (float*)d_out;
    (void)in_sizes; (void)n_in; (void)out_size; (void)d_ws; (void)ws_size;

    const size_t shmem =
        (size_t)(MT * KTOT + KC * GP + MT * Gc + MT * Hc) * sizeof(float); // ~185 KB

    // CDNA5 WGP has 320 KB LDS; raise the dynamic-LDS cap (idempotent, capture-safe)
    (void)hipFuncSetAttribute((const void*)onlstm_fused_kernel,
                              hipFuncAttributeMaxDynamicSharedMemorySize,
                              (int)shmem);

    dim3 grid((Tc / MT) * 2);   // 128 row tiles x 2 directions = 256 blocks
    dim3 block(BLOCK);          // 8 wave32s
    onlstm_fused_kernel<<<grid, block, shmem, stream>>>(x, sl, W_ih, b_ih,
                                                        W_hh, b_hh, out);
}